// ImageGCN_21320217657492
// MI455X (gfx1250) — compile-verified
//
#include <hip/hip_runtime.h>

typedef float v2f __attribute__((ext_vector_type(2)));
typedef float v8f __attribute__((ext_vector_type(8)));

#define N_NODES 30000
#define E_EDGES 480000
#define IN_DIM  256
#define REL_DIM 128
#define OUT_DIM 512
#define SUM_DIM (IN_DIM + REL_DIM)       // 384
#define CAT_DIM (2 * IN_DIM + REL_DIM)   // 640

// ---------------------------------------------------------------------------
// Wave-level 16x16 fp32 WMMA tile GEMM helper (layouts per CDNA5 ISA 7.12.2).
// ---------------------------------------------------------------------------
__device__ __forceinline__ v8f wmma_tile_f32(const float* __restrict__ A, int lda,
                                             const float* __restrict__ B, int ldb,
                                             int K, v8f acc) {
    const int lane  = threadIdx.x & 31;
    const int m     = lane & 15;            // M for A-frag, N for B-frag
    const int khalf = (lane >> 4) << 1;     // 0 or 2
#pragma unroll 4
    for (int k0 = 0; k0 < K; k0 += 4) {
        v2f a = *(const v2f*)(A + m * lda + k0 + khalf);
        const float* bp = B + (size_t)(k0 + khalf) * ldb + m;
        v2f b;
        b.x = bp[0];
        b.y = bp[ldb];
        acc = __builtin_amdgcn_wmma_f32_16x16x4_f32(
            /*neg_a=*/false, a, /*neg_b=*/false, b,
            /*c_mod=*/(short)0, acc, /*reuse_a=*/false, /*reuse_b=*/false);
    }
    return acc;
}

// ---------------------------------------------------------------------------
// Stage 0: in-degree histogram (1 int atomic per edge)
// ---------------------------------------------------------------------------
__global__ void deg_kernel(const int* __restrict__ dst, int* __restrict__ deg) {
    const int e = blockIdx.x * 256 + threadIdx.x;
    if (e < E_EDGES) atomicAdd(&deg[dst[e]], 1);
}

// ---------------------------------------------------------------------------
// Stage 1: h2 = h @ Wn + bn, fused self-scatter term:
//          hsum[r][0:256] = node_att[r] * deg[r] * h2[r]   (plain store onto
//          zeroed hsum; equals the  Σ_{e:dst=r} att[r]*h2[r]  message half)
// grid (IN_DIM/64, N_NODES/16), block 128 (4 waves)
// ---------------------------------------------------------------------------
__global__ void gemm_node_kernel(const float* __restrict__ h,
                                 const float* __restrict__ Wn,
                                 const float* __restrict__ bn,
                                 const float* __restrict__ node_att,
                                 const int* __restrict__ deg,
                                 float* __restrict__ h2,
                                 float* __restrict__ hsum) {
    const int wave = threadIdx.x >> 5;
    const int lane = threadIdx.x & 31;
    const int col0 = (blockIdx.x * 4 + wave) * 16;
    const int row0 = blockIdx.y * 16;

    v8f acc = {};
    acc = wmma_tile_f32(h + (size_t)row0 * IN_DIM, IN_DIM,
                        Wn + col0, IN_DIM, IN_DIM, acc);

    const int n    = col0 + (lane & 15);
    const float bb = bn[n];
    const int rhi  = (lane >> 4) << 3;       // lanes 16-31 hold M = i+8
#pragma unroll
    for (int i = 0; i < 8; ++i) {
        const int r   = row0 + i + rhi;
        const float v = acc[i] + bb;
        h2[(size_t)r * IN_DIM + n] = v;
        const float scale = node_att[r] * (float)deg[r];
        hsum[(size_t)r * SUM_DIM + n] = scale * v;
    }
}

// ---------------------------------------------------------------------------
// Stage 2: z2 = edge_att * (rel @ Wr + br), scatter-add into hsum[dst][256+n]
// grid (REL_DIM/64, E_EDGES/16), block 128
// ---------------------------------------------------------------------------
__global__ void gemm_rel_scatter_kernel(const float* __restrict__ rel,
                                        const float* __restrict__ Wr,
                                        const float* __restrict__ br,
                                        const float* __restrict__ edge_att,
                                        const int* __restrict__ dst,
                                        float* __restrict__ hsum) {
    const int wave = threadIdx.x >> 5;
    const int lane = threadIdx.x & 31;
    const int col0 = (blockIdx.x * 4 + wave) * 16;
    const int e0   = blockIdx.y * 16;

    v8f acc = {};
    acc = wmma_tile_f32(rel + (size_t)e0 * REL_DIM, REL_DIM,
                        Wr + col0, REL_DIM, REL_DIM, acc);

    const int n    = col0 + (lane & 15);
    const float bb = br[n];
    const int rhi  = (lane >> 4) << 3;
#pragma unroll
    for (int i = 0; i < 8; ++i) {
        const int e   = e0 + i + rhi;
        const float v = (acc[i] + bb) * edge_att[e];
        const int d   = dst[e];
        atomicAdd(&hsum[(size_t)d * SUM_DIM + IN_DIM + n], v);
    }
}

// ---------------------------------------------------------------------------
// Stage 3: scatter only the src half:  hsum[dst][0:256] += att[src]*h2[src]
// (dst half already handled analytically via the degree term in stage 1)
// One wave per edge; 32 lanes x 2 float4 cover 256 columns.
// grid (E_EDGES/8), block 256 (8 waves)
// ---------------------------------------------------------------------------
__global__ void edge_src_scatter_kernel(const float* __restrict__ h2,
                                        const float* __restrict__ node_att,
                                        const int* __restrict__ src,
                                        const int* __restrict__ dst,
                                        float* __restrict__ hsum) {
    const int wave = threadIdx.x >> 5;
    const int lane = threadIdx.x & 31;
    const int e    = blockIdx.x * 8 + wave;
    if (e >= E_EDGES) return;

    const int s    = src[e];
    const int d    = dst[e];
    const float as = node_att[s];

    const float4* hs = (const float4*)(h2 + (size_t)s * IN_DIM);
    float* outrow    = hsum + (size_t)d * SUM_DIM;

    // gfx1250 global_prefetch_b8: pull gather row + scatter target toward L2
    __builtin_prefetch(hs, 0, 1);
    __builtin_prefetch(outrow, 1, 1);

#pragma unroll
    for (int i = 0; i < 2; ++i) {
        const int idx = lane + 32 * i;       // float4 index 0..63
        const float4 a = hs[idx];
        const int c = idx * 4;
        atomicAdd(&outrow[c + 0], as * a.x);
        atomicAdd(&outrow[c + 1], as * a.y);
        atomicAdd(&outrow[c + 2], as * a.z);
        atomicAdd(&outrow[c + 3], as * a.w);
    }
}

// ---------------------------------------------------------------------------
// Stage 4: out = relu(concat(h2, hsum) @ Wa + ba)   [30000 x 512], K = 640
// Concat handled in the A-fragment load: 4-chunks never straddle 256.
// grid (OUT_DIM/64, N_NODES/16), block 128
// ---------------------------------------------------------------------------
__global__ void gemm_out_kernel(const float* __restrict__ h2,
                                const float* __restrict__ hsum,
                                const float* __restrict__ Wa,
                                const float* __restrict__ ba,
                                float* __restrict__ out) {
    const int wave  = threadIdx.x >> 5;
    const int lane  = threadIdx.x & 31;
    const int col0  = (blockIdx.x * 4 + wave) * 16;
    const int row0  = blockIdx.y * 16;
    const int m     = lane & 15;
    const int khalf = (lane >> 4) << 1;

    const float* Ah = h2   + ((size_t)row0 + m) * IN_DIM;
    const float* As = hsum + ((size_t)row0 + m) * SUM_DIM;

    v8f acc = {};
#pragma unroll 4
    for (int k0 = 0; k0 < CAT_DIM; k0 += 4) {
        v2f a;
        if (k0 < IN_DIM) {
            a = *(const v2f*)(Ah + k0 + khalf);
        } else {
            a = *(const v2f*)(As + (k0 - IN_DIM) + khalf);
        }
        const float* bp = Wa + (size_t)(k0 + khalf) * OUT_DIM + col0 + m;
        v2f b;
        b.x = bp[0];
        b.y = bp[OUT_DIM];
        acc = __builtin_amdgcn_wmma_f32_16x16x4_f32(
            false, a, false, b, (short)0, acc, false, false);
    }

    const int n    = col0 + m;
    const float bb = ba[n];
    const int rhi  = (lane >> 4) << 3;
#pragma unroll
    for (int i = 0; i < 8; ++i) {
        const int r = row0 + i + rhi;
        out[(size_t)r * OUT_DIM + n] = fmaxf(acc[i] + bb, 0.0f);
    }
}

// ---------------------------------------------------------------------------
// Launch: memsets -> deg -> node GEMM (fused self term) -> rel GEMM+scatter
//         -> src scatter -> out GEMM.  Stream order gives all dependencies.
// ---------------------------------------------------------------------------
extern "C" void kernel_launch(void* const* d_in, const int* in_sizes, int n_in,
                              void* d_out, int out_size, void* d_ws, size_t ws_size,
                              hipStream_t stream) {
    const float* h        = (const float*)d_in[0];
    const float* node_att = (const float*)d_in[1];
    const float* rel      = (const float*)d_in[2];
    const float* edge_att = (const float*)d_in[3];
    const float* Wn       = (const float*)d_in[4];
    const float* bn       = (const float*)d_in[5];
    const float* Wr       = (const float*)d_in[6];
    const float* br       = (const float*)d_in[7];
    const float* Wa       = (const float*)d_in[8];
    const float* ba       = (const float*)d_in[9];
    const int*   src      = (const int*)d_in[10];
    const int*   dst      = (const int*)d_in[11];
    float* out = (float*)d_out;

    float* h2   = (float*)d_ws;                        // 30000*256 f32 = 30.72 MB
    float* hsum = h2 + (size_t)N_NODES * IN_DIM;       // 30000*384 f32 = 46.08 MB
    int*   deg  = (int*)(hsum + (size_t)N_NODES * SUM_DIM); // 30000 i32 = 0.12 MB

    hipMemsetAsync(hsum, 0, (size_t)N_NODES * SUM_DIM * sizeof(float), stream);
    hipMemsetAsync(deg, 0, (size_t)N_NODES * sizeof(int), stream);

    deg_kernel<<<dim3((E_EDGES + 255) / 256), 256, 0, stream>>>(dst, deg);

    gemm_node_kernel<<<dim3(IN_DIM / 64, N_NODES / 16), 128, 0, stream>>>(
        h, Wn, bn, node_att, deg, h2, hsum);

    gemm_rel_scatter_kernel<<<dim3(REL_DIM / 64, E_EDGES / 16), 128, 0, stream>>>(
        rel, Wr, br, edge_att, dst, hsum);

    edge_src_scatter_kernel<<<dim3(E_EDGES / 8), 256, 0, stream>>>(
        h2, node_att, src, dst, hsum);

    gemm_out_kernel<<<dim3(OUT_DIM / 64, N_NODES / 16), 128, 0, stream>>>(
        h2, hsum, Wa, ba, out);
}